// SurfEvalTorch_575525617695
// MI455X (gfx1250) — compile-verified
//
#include <hip/hip_runtime.h>

typedef __attribute__((ext_vector_type(2))) float v2f;
typedef __attribute__((ext_vector_type(8))) float v8f;

#define NUM_CTRL 16
#define OUT_U    32
#define OUT_V    128
#define BATCH    512

// LDS row stride 18 floats: even (8B-aligned v2f fragment loads) and
// row*18 mod 64 hits 16 distinct banks for 16 consecutive rows.
#define A_STR 18
#define B_STR 18
#define T_STR 18
#define T_PLANE (OUT_U * T_STR)   // 576

__global__ __launch_bounds__(256)
void nurbs_wmma_kernel(const float* __restrict__ ctrl,
                       const float* __restrict__ Nu,
                       const float* __restrict__ Nv,
                       const int*   __restrict__ uspan,
                       const int*   __restrict__ vspan,
                       float*       __restrict__ out)
{
    __shared__ __align__(16) float As[OUT_U * A_STR];   // dense Nu expansion, 32x16 (pad 18)
    __shared__ __align__(16) float Bs[OUT_V * B_STR];   // dense Nv expansion, 128x16 (pad 18)
    __shared__ __align__(16) float Cs[NUM_CTRL * 64];   // ctrl tile [k][n][d] == memory order
    __shared__ __align__(16) float Ts[4 * T_PLANE];     // T_d (32x16) per channel d

    const int tid  = threadIdx.x;
    const int b    = blockIdx.x;
    const int lane = tid & 31;
    const int wave = tid >> 5;
    const int ln   = lane & 15;          // N / M position within half-wave
    const int hi   = (lane >> 4) & 1;    // half-wave selector (K offset +2 / M offset +8)
    const int koff = hi * 2;             // K base offset for this half-wave

    // ---- phase 0: zero dense basis mats, copy ctrl tile (float4, 4KB) ----
    for (int i = tid; i < OUT_U * A_STR; i += 256) As[i] = 0.0f;
    for (int i = tid; i < OUT_V * B_STR; i += 256) Bs[i] = 0.0f;
    {
        const float4* src = (const float4*)(ctrl + (size_t)b * (NUM_CTRL * NUM_CTRL * 4));
        float4*       dst = (float4*)Cs;
        dst[tid] = src[tid];
    }
    __syncthreads();

    // ---- phase 1: scatter Nu/Nv into dense rows (4 nonzeros each) ----
    if (tid < OUT_U) {
        int s = uspan[tid] - 3;
        #pragma unroll
        for (int i = 0; i < 4; ++i) As[tid * A_STR + s + i] = Nu[tid * 4 + i];
    }
    if (tid >= 128) {                     // waves 4-7 handle the 128 v rows
        int t = tid - 128;
        int s = vspan[t] - 3;
        #pragma unroll
        for (int j = 0; j < 4; ++j) Bs[t * B_STR + s + j] = Nv[t * 4 + j];
    }
    __syncthreads();

    // ---- stage 1: T_d(32x16) = A(32x16) @ C_d(16x16); 8 wave jobs ----
    {
        const int mt = wave & 1;          // u tile
        const int d  = wave >> 1;         // channel
        const int m0 = mt * 16;
        v8f acc = {};
        #pragma unroll
        for (int k0 = 0; k0 < 16; k0 += 4) {
            // A operand 16x4: one aligned b64 per lane (K pair is contiguous)
            v2f a = *(const v2f*)&As[(m0 + ln) * A_STR + k0 + koff];
            // B operand 4x16: element (k,n) = C_d[k][n] = Cs[k*64 + n*4 + d] (not contiguous)
            v2f bb;
            bb.x = Cs[(k0 + koff + 0) * 64 + ln * 4 + d];
            bb.y = Cs[(k0 + koff + 1) * 64 + ln * 4 + d];
            acc = __builtin_amdgcn_wmma_f32_16x16x4_f32(false, a, false, bb,
                                                        (short)0, acc, false, false);
        }
        #pragma unroll
        for (int r = 0; r < 8; ++r) {
            int row = m0 + r + (hi ? 8 : 0);
            Ts[d * T_PLANE + row * T_STR + ln] = acc[r];
        }
    }
    __syncthreads();

    // ---- stage 2: S_d(32x128) = T_d @ B^T; each wave does 2 (mt,nt) tiles,
    //      all 4 channels at once so xyz/w happens in registers ----
    float* obase = out + (size_t)b * (OUT_U * OUT_V * 3);
    #pragma unroll
    for (int jj = 0; jj < 2; ++jj) {
        const int job = wave + jj * 8;    // 0..15
        const int mt  = job & 1;
        const int nt  = job >> 1;         // 0..7
        const int m0  = mt * 16;
        const int n0  = nt * 16;
        v8f c0 = {}, c1 = {}, c2 = {}, c3 = {};
        #pragma unroll
        for (int k0 = 0; k0 < 16; k0 += 4) {
            // B operand 4x16: element (k,n) = Bdense[n0+n][k0+k]; K pair contiguous -> b64
            v2f bb = *(const v2f*)&Bs[(n0 + ln) * B_STR + k0 + koff];
            const int kx = (m0 + ln) * T_STR + k0 + koff;
            v2f a0 = *(const v2f*)&Ts[0 * T_PLANE + kx];
            v2f a1 = *(const v2f*)&Ts[1 * T_PLANE + kx];
            v2f a2 = *(const v2f*)&Ts[2 * T_PLANE + kx];
            v2f a3 = *(const v2f*)&Ts[3 * T_PLANE + kx];
            c0 = __builtin_amdgcn_wmma_f32_16x16x4_f32(false, a0, false, bb, (short)0, c0, false, false);
            c1 = __builtin_amdgcn_wmma_f32_16x16x4_f32(false, a1, false, bb, (short)0, c1, false, false);
            c2 = __builtin_amdgcn_wmma_f32_16x16x4_f32(false, a2, false, bb, (short)0, c2, false, false);
            c3 = __builtin_amdgcn_wmma_f32_16x16x4_f32(false, a3, false, bb, (short)0, c3, false, false);
        }
        // rational divide + streaming store (lane L reg r <-> M=m0+r+8*hi, N=n0+ln)
        #pragma unroll
        for (int r = 0; r < 8; ++r) {
            int row = m0 + r + (hi ? 8 : 0);
            int col = n0 + ln;
            float w   = fmaxf(c3[r], 1e-8f);
            float inv = 1.0f / w;
            size_t o = ((size_t)row * OUT_V + col) * 3;
            obase[o + 0] = c0[r] * inv;
            obase[o + 1] = c1[r] * inv;
            obase[o + 2] = c2[r] * inv;
        }
    }
}

extern "C" void kernel_launch(void* const* d_in, const int* in_sizes, int n_in,
                              void* d_out, int out_size, void* d_ws, size_t ws_size,
                              hipStream_t stream) {
    const float* ctrl  = (const float*)d_in[0];
    const float* Nu    = (const float*)d_in[1];
    const float* Nv    = (const float*)d_in[2];
    const int*   uspan = (const int*)d_in[3];
    const int*   vspan = (const int*)d_in[4];
    float*       out   = (float*)d_out;

    nurbs_wmma_kernel<<<dim3(BATCH), dim3(256), 0, stream>>>(ctrl, Nu, Nv, uspan, vspan, out);
}